// APPNP_18047452578189
// MI455X (gfx1250) — compile-verified
//
#include <hip/hip_runtime.h>
#include <hip/hip_bf16.h>
#include <stdint.h>

typedef __attribute__((ext_vector_type(2))) float v2f;
typedef __attribute__((ext_vector_type(8))) float v8f;
typedef long long i64;

#define N_NODES 100000
#define C 16
#define F 512

// ---------------------------------------------------------------------------
// Zero a float buffer (grid-stride)
// ---------------------------------------------------------------------------
__global__ void zero_f32_kernel(float* __restrict__ p, size_t n) {
    size_t i = (size_t)blockIdx.x * blockDim.x + threadIdx.x;
    size_t stride = (size_t)gridDim.x * blockDim.x;
    for (; i < n; i += stride) p[i] = 0.0f;
}

// ---------------------------------------------------------------------------
// Degree accumulation: deg_out[src[e]] += 1, deg_in[dst[e]] += 1
// ---------------------------------------------------------------------------
__global__ void degree_kernel(const i64* __restrict__ src, const i64* __restrict__ dst,
                              float* __restrict__ deg_out, float* __restrict__ deg_in,
                              int n_edges) {
    int e = blockIdx.x * blockDim.x + threadIdx.x;
    if (e < n_edges) {
        atomicAdd(&deg_out[(int)src[e]], 1.0f);
        atomicAdd(&deg_in[(int)dst[e]], 1.0f);
    }
}

// ---------------------------------------------------------------------------
// deg -> clip(deg,1)^-0.5 in place (covers both norm arrays: length 2N)
// ---------------------------------------------------------------------------
__global__ void norm_kernel(float* __restrict__ deg, int n) {
    int i = blockIdx.x * blockDim.x + threadIdx.x;
    if (i < n) deg[i] = rsqrtf(fmaxf(deg[i], 1.0f));
}

// ---------------------------------------------------------------------------
// WMMA GEMM: h0[n, c] = sum_f x[n, f] * W[c, f] + b[c]
// One wave per 16-row tile, V_WMMA_F32_16X16X4_F32, 128 k-steps of 4.
//
// A (16x4 f32) lane layout: lanes 0-15 hold M=lane, K={k,k+1} in vgpr0/1;
//                           lanes 16-31 hold M=lane-16, K={k+2,k+3}.
// B (4x16 f32) lane layout: lanes 0-15 hold N=lane, K={k,k+1};
//                           lanes 16-31 hold N=lane-16, K={k+2,k+3}.
// C/D (16x16 f32): vgpr v, lane l -> (M = v + 8*(l>>4), N = l&15).
// ---------------------------------------------------------------------------
__global__ void gemm_wmma_kernel(const float* __restrict__ x,
                                 const float* __restrict__ W,
                                 const float* __restrict__ bias,
                                 float* __restrict__ h0,
                                 int n_nodes) {
    const int wave_id = (blockIdx.x * blockDim.x + threadIdx.x) >> 5;
    const int lane    = threadIdx.x & 31;
    const int row0    = wave_id * 16;
    if (row0 >= n_nodes) return;   // whole-wave uniform: EXEC stays all-ones

    const int half = lane >> 4;    // 0: K pair {0,1}, 1: K pair {2,3}
    const int m    = lane & 15;    // row (A) / col (B) owned by this lane

    const float* __restrict__ xrow = x + (size_t)(row0 + m) * F;
    const float* __restrict__ wrow = W + (size_t)m * F;

    v8f acc = {};
    #pragma unroll 4
    for (int k = 0; k < F; k += 4) {
        const int kk = k + 2 * half;
        v2f a, bm;
        a.x  = xrow[kk];  a.y  = xrow[kk + 1];
        bm.x = wrow[kk];  bm.y = wrow[kk + 1];
        acc = __builtin_amdgcn_wmma_f32_16x16x4_f32(
            /*neg_a=*/false, a, /*neg_b=*/false, bm,
            /*c_mod=*/(short)0, acc, /*reuse_a=*/false, /*reuse_b=*/false);
    }

    const float bv = bias[m];      // N = m for every C/D vgpr of this lane
    #pragma unroll
    for (int v = 0; v < 8; ++v) {
        const int r = row0 + v + 8 * half;
        h0[(size_t)r * C + m] = acc[v] + bv;
    }
}

// ---------------------------------------------------------------------------
// Scatter: agg[dst[e], c] += h[src[e], c] * norm_out[src[e]]
// 16 lanes per edge (one per channel) -> coalesced 64B gather per edge,
// fire-and-forget global_atomic_add_f32 into L2-resident agg (6.4 MB).
// ---------------------------------------------------------------------------
__global__ void scatter_kernel(const float* __restrict__ h,
                               const float* __restrict__ norm_out,
                               const i64* __restrict__ src,
                               const i64* __restrict__ dst,
                               float* __restrict__ agg,
                               int n_edges) {
    const long long t = (long long)blockIdx.x * blockDim.x + threadIdx.x;
    const long long e = t >> 4;
    const int c = (int)(t & 15);
    if (e >= n_edges) return;
    const int s = (int)src[e];
    const int d = (int)dst[e];
    const float v = h[(size_t)s * C + c] * norm_out[s];
    atomicAdd(&agg[(size_t)d * C + c], v);
}

// ---------------------------------------------------------------------------
// Blend (in place on agg): out[t] = 0.5 * agg[t] * norm_in[node] + 0.5 * h0[t]
// ---------------------------------------------------------------------------
__global__ void combine_kernel(float* __restrict__ agg,
                               const float* __restrict__ norm_in,
                               const float* __restrict__ h0,
                               int n_nodes) {
    const int t = blockIdx.x * blockDim.x + threadIdx.x;
    const int node = t >> 4;
    if (node >= n_nodes) return;
    agg[t] = 0.5f * (agg[t] * norm_in[node]) + 0.5f * h0[t];
}

// ---------------------------------------------------------------------------
// Launch
// ---------------------------------------------------------------------------
extern "C" void kernel_launch(void* const* d_in, const int* in_sizes, int n_in,
                              void* d_out, int out_size, void* d_ws, size_t ws_size,
                              hipStream_t stream) {
    const float* x    = (const float*)d_in[0];   // [N, 512]
    const float* W    = (const float*)d_in[1];   // [16, 512]
    const float* bias = (const float*)d_in[2];   // [16]
    const i64*   src  = (const i64*)d_in[3];     // [E]
    const i64*   dst  = (const i64*)d_in[4];     // [E]
    float* out = (float*)d_out;                  // [N, 16]
    const int n_edges = in_sizes[3];

    // Workspace layout
    char* ws = (char*)d_ws;
    float* norm_out = (float*)ws;                               // N floats
    float* norm_in  = norm_out + N_NODES;                       // N floats
    float* h0       = norm_in + N_NODES;                        // N*C floats
    float* buf1     = h0 + (size_t)N_NODES * C;                 // N*C floats
    float* buf2     = buf1 + (size_t)N_NODES * C;               // N*C floats

    const int NT = 256;

    // 1) degrees -> norms
    zero_f32_kernel<<<2048, NT, 0, stream>>>(norm_out, (size_t)2 * N_NODES);
    degree_kernel<<<(n_edges + NT - 1) / NT, NT, 0, stream>>>(src, dst, norm_out, norm_in, n_edges);
    norm_kernel<<<(2 * N_NODES + NT - 1) / NT, NT, 0, stream>>>(norm_out, 2 * N_NODES);

    // 2) h0 = x @ W^T + b   (WMMA, one wave / 16-row tile, 8 waves / block)
    {
        const int tiles = (N_NODES + 15) / 16;                  // 6250
        const int waves_per_block = NT / 32;                    // 8
        const int blocks = (tiles + waves_per_block - 1) / waves_per_block;
        gemm_wmma_kernel<<<blocks, NT, 0, stream>>>(x, W, bias, h0, N_NODES);
    }

    // 3) K=3 propagation steps; blend runs in place on the agg buffer.
    //    iter1: h0  -> buf1 ; iter2: buf1 -> buf2 ; iter3: buf2 -> d_out
    const long long sthreads = (long long)n_edges * C;
    const int sblocks = (int)((sthreads + NT - 1) / NT);
    const int cblocks = (N_NODES * C + NT - 1) / NT;

    const float* h_prev = h0;
    float* aggs[3] = { buf1, buf2, out };
    for (int it = 0; it < 3; ++it) {
        float* agg = aggs[it];
        zero_f32_kernel<<<2048, NT, 0, stream>>>(agg, (size_t)N_NODES * C);
        scatter_kernel<<<sblocks, NT, 0, stream>>>(h_prev, norm_out, src, dst, agg, n_edges);
        combine_kernel<<<cblocks, NT, 0, stream>>>(agg, norm_in, h0, N_NODES);
        h_prev = agg;
    }
}